// MarcusGATConv_8572754723150
// MI455X (gfx1250) — compile-verified
//
#include <hip/hip_runtime.h>
#include <math.h>

// CDNA5 / gfx1250: wave32, WMMA f32 16x16x4.
typedef __attribute__((ext_vector_type(2))) float v2f;
typedef __attribute__((ext_vector_type(8))) float v8f;

#define D_FEAT 64

// ---------------------------------------------------------------------------
// Kernel 0: initialize segment-max to a large negative, denom and out to zero.
// ---------------------------------------------------------------------------
__global__ void gat_init_kernel(float* __restrict__ smax,
                                float* __restrict__ denom,
                                float* __restrict__ out, int n) {
    int i = blockIdx.x * blockDim.x + threadIdx.x;
    if (i < n) {
        smax[i]  = -3.0e38f;   // finite "neutral" max (reference neutralizes -inf)
        denom[i] = 0.0f;
        out[i]   = 0.0f;
    }
}

// ---------------------------------------------------------------------------
// Kernel 1: feat_src = relu(h_src @ W^T + b) via V_WMMA_F32_16X16X4_F32.
// One wave owns a 16-row strip and all 64 output columns (4 16x16 D tiles).
// A 16x4 f32 operand layout (ISA 7.12.2): lanes 0-15 carry K={k,k+1},
// lanes 16-31 carry K={k+2,k+3}; B (=W^T) mirrored; C/D: VGPR r holds rows
// r (lanes 0-15) and r+8 (lanes 16-31).
// ---------------------------------------------------------------------------
__global__ void __launch_bounds__(128)
gat_feat_gemm_relu_wmma(const float* __restrict__ A,   // h_src [N,64]
                        const float* __restrict__ W,   // W_src [64,64] (row j = output col j)
                        const float* __restrict__ bias,
                        float* __restrict__ out,       // feat_src [N,64]
                        int N) {
    const int lane = threadIdx.x & 31;
    const int wave = threadIdx.x >> 5;
    const int tile = blockIdx.x * 4 + wave;     // 16-row tile id
    const int row0 = tile * 16;
    if (row0 >= N) return;                      // wave-uniform: EXEC stays all-ones

    const int half = lane >> 4;                 // 0: K pair {0,1}; 1: K pair {2,3}
    const int lid  = lane & 15;

    const float* Arow = A + (size_t)(row0 + lid) * D_FEAT;

    v8f acc0 = {}, acc1 = {}, acc2 = {}, acc3 = {};

    #pragma unroll 4
    for (int s = 0; s < 16; ++s) {
        const int k = s * 4 + half * 2;
        v2f a;
        a.x = Arow[k];
        a.y = Arow[k + 1];
        // B = W^T, so B[k][n] = W[n][k]; column tile t covers n = 16t + lid.
        const float* w0 = W + (size_t)lid * D_FEAT + k;
        const float* w1 = w0 + 16 * D_FEAT;
        const float* w2 = w0 + 32 * D_FEAT;
        const float* w3 = w0 + 48 * D_FEAT;
        v2f b0, b1, b2, b3;
        b0.x = w0[0]; b0.y = w0[1];
        b1.x = w1[0]; b1.y = w1[1];
        b2.x = w2[0]; b2.y = w2[1];
        b3.x = w3[0]; b3.y = w3[1];
        acc0 = __builtin_amdgcn_wmma_f32_16x16x4_f32(false, a, false, b0, (short)0, acc0, false, false);
        acc1 = __builtin_amdgcn_wmma_f32_16x16x4_f32(false, a, false, b1, (short)0, acc1, false, false);
        acc2 = __builtin_amdgcn_wmma_f32_16x16x4_f32(false, a, false, b2, (short)0, acc2, false, false);
        acc3 = __builtin_amdgcn_wmma_f32_16x16x4_f32(false, a, false, b3, (short)0, acc3, false, false);
    }

    const float bv0 = bias[ 0 + lid];
    const float bv1 = bias[16 + lid];
    const float bv2 = bias[32 + lid];
    const float bv3 = bias[48 + lid];

    #pragma unroll
    for (int r = 0; r < 8; ++r) {
        const int m = row0 + r + 8 * half;     // D layout: VGPR r -> rows r / r+8
        float* orow = out + (size_t)m * D_FEAT + lid;
        orow[ 0] = fmaxf(acc0[r] + bv0, 0.0f);
        orow[16] = fmaxf(acc1[r] + bv1, 0.0f);
        orow[32] = fmaxf(acc2[r] + bv2, 0.0f);
        orow[48] = fmaxf(acc3[r] + bv3, 0.0f);
    }
}

// ---------------------------------------------------------------------------
// Edge pass mapping: 16 threads per edge, each owns a contiguous float4
// feature slice -> coalesced global_load_b128 gathers + 4 f32 atomics.
// ---------------------------------------------------------------------------
__global__ void __launch_bounds__(256)
gat_edge_max_kernel(const float* __restrict__ h_src, const float* __restrict__ h_dst,
                    const int* __restrict__ src_idx, const int* __restrict__ dst_idx,
                    float* __restrict__ smax, int E) {
    const int t = blockIdx.x * blockDim.x + threadIdx.x;
    const int e = t >> 4;
    if (e >= E) return;
    const int q = (t & 15) << 2;
    const int s = src_idx[e];
    const int d = dst_idx[e];
    const float4 hs = *reinterpret_cast<const float4*>(h_src + (size_t)s * D_FEAT + q);
    const float4 hd = *reinterpret_cast<const float4*>(h_dst + (size_t)d * D_FEAT + q);
    float* p = smax + (size_t)d * D_FEAT + q;
    atomicMax(p + 0, hs.x * hd.x);   // global_atomic_max_num_f32
    atomicMax(p + 1, hs.y * hd.y);
    atomicMax(p + 2, hs.z * hd.z);
    atomicMax(p + 3, hs.w * hd.w);
}

__global__ void __launch_bounds__(256)
gat_edge_expsum_kernel(const float* __restrict__ h_src, const float* __restrict__ h_dst,
                       const int* __restrict__ src_idx, const int* __restrict__ dst_idx,
                       const float* __restrict__ smax, float* __restrict__ denom, int E) {
    const int t = blockIdx.x * blockDim.x + threadIdx.x;
    const int e = t >> 4;
    if (e >= E) return;
    const int q = (t & 15) << 2;
    const int s = src_idx[e];
    const int d = dst_idx[e];
    const float4 hs = *reinterpret_cast<const float4*>(h_src + (size_t)s * D_FEAT + q);
    const float4 hd = *reinterpret_cast<const float4*>(h_dst + (size_t)d * D_FEAT + q);
    const float4 mx = *reinterpret_cast<const float4*>(smax  + (size_t)d * D_FEAT + q);
    float* p = denom + (size_t)d * D_FEAT + q;
    atomicAdd(p + 0, __expf(hs.x * hd.x - mx.x));  // global_atomic_add_f32
    atomicAdd(p + 1, __expf(hs.y * hd.y - mx.y));
    atomicAdd(p + 2, __expf(hs.z * hd.z - mx.z));
    atomicAdd(p + 3, __expf(hs.w * hd.w - mx.w));
}

__global__ void __launch_bounds__(256)
gat_edge_aggregate_kernel(const float* __restrict__ h_src, const float* __restrict__ h_dst,
                          const int* __restrict__ src_idx, const int* __restrict__ dst_idx,
                          const float* __restrict__ smax, const float* __restrict__ denom,
                          const float* __restrict__ feat, float* __restrict__ out, int E) {
    const int t = blockIdx.x * blockDim.x + threadIdx.x;
    const int e = t >> 4;
    if (e >= E) return;
    const int q = (t & 15) << 2;
    const int s = src_idx[e];
    const int d = dst_idx[e];
    const float4 hs = *reinterpret_cast<const float4*>(h_src + (size_t)s * D_FEAT + q);
    const float4 hd = *reinterpret_cast<const float4*>(h_dst + (size_t)d * D_FEAT + q);
    const float4 mx = *reinterpret_cast<const float4*>(smax  + (size_t)d * D_FEAT + q);
    const float4 dn = *reinterpret_cast<const float4*>(denom + (size_t)d * D_FEAT + q);
    const float4 fs = *reinterpret_cast<const float4*>(feat  + (size_t)s * D_FEAT + q);
    float* p = out + (size_t)d * D_FEAT + q;
    atomicAdd(p + 0, fs.x * __expf(hs.x * hd.x - mx.x) / dn.x);
    atomicAdd(p + 1, fs.y * __expf(hs.y * hd.y - mx.y) / dn.y);
    atomicAdd(p + 2, fs.z * __expf(hs.z * hd.z - mx.z) / dn.z);
    atomicAdd(p + 3, fs.w * __expf(hs.w * hd.w - mx.w) / dn.w);
}

// ---------------------------------------------------------------------------
// Launcher
// ---------------------------------------------------------------------------
extern "C" void kernel_launch(void* const* d_in, const int* in_sizes, int n_in,
                              void* d_out, int out_size, void* d_ws, size_t ws_size,
                              hipStream_t stream) {
    const float* h_src   = (const float*)d_in[0];   // [N_src, 64]
    const float* h_dst   = (const float*)d_in[1];   // [N_dst, 64]
    const int*   src_idx = (const int*)  d_in[2];   // [E]
    const int*   dst_idx = (const int*)  d_in[3];   // [E]
    const float* W_src   = (const float*)d_in[4];   // [64, 64]
    const float* b_src   = (const float*)d_in[5];   // [64]

    const int n_src = in_sizes[0] / D_FEAT;
    const int n_dst = in_sizes[1] / D_FEAT;
    const int E     = in_sizes[2];

    float* out   = (float*)d_out;                       // [N_dst, 1, 64] flat
    float* feat  = (float*)d_ws;                        // [N_src, 64]
    float* smax  = feat  + (size_t)n_src * D_FEAT;      // [N_dst, 64]
    float* denom = smax  + (size_t)n_dst * D_FEAT;      // [N_dst, 64]

    // 0) init segment buffers + output
    const int nd = n_dst * D_FEAT;
    gat_init_kernel<<<(nd + 255) / 256, 256, 0, stream>>>(smax, denom, out, nd);

    // 1) feat_src = relu(h_src @ W^T + b) — WMMA f32 16x16x4
    const int tiles = (n_src + 15) / 16;
    gat_feat_gemm_relu_wmma<<<(tiles + 3) / 4, 128, 0, stream>>>(h_src, W_src, b_src, feat, n_src);

    // 2-4) edge passes: 16 threads per edge
    const long long et = (long long)E * 16;
    const int eb = (int)((et + 255) / 256);
    gat_edge_max_kernel<<<eb, 256, 0, stream>>>(h_src, h_dst, src_idx, dst_idx, smax, E);
    gat_edge_expsum_kernel<<<eb, 256, 0, stream>>>(h_src, h_dst, src_idx, dst_idx, smax, denom, E);
    gat_edge_aggregate_kernel<<<eb, 256, 0, stream>>>(h_src, h_dst, src_idx, dst_idx,
                                                      smax, denom, feat, out, E);
}